// HBNS_40346922779262
// MI455X (gfx1250) — compile-verified
//
#include <hip/hip_runtime.h>
#include <hip/hip_bf16.h>

typedef __attribute__((ext_vector_type(2))) float v2f;
typedef __attribute__((ext_vector_type(8))) float v8f;

#define DHID 256
#define NEG_SLOPE 0.2f
// LDS row stride (dwords) for transposed W: 260 ≡ 4 (mod 64 banks) makes the
// two half-wave b64 B-fragment reads land in 64 distinct banks, and keeps
// 8-byte alignment for ds_load_b64 (260 is even, k0 is even).
#define WT_STRIDE 260
#define WT_LDS_BYTES (DHID * WT_STRIDE * 4)

__device__ __forceinline__ int imin(int a, int b) { return a < b ? a : b; }

// ---------------------------------------------------------------------------
// Zero-fill (d_out / accumulators are poisoned by the harness)
// ---------------------------------------------------------------------------
__global__ __launch_bounds__(256) void HBNS_fill_zero(float* __restrict__ p, long long n)
{
    long long i = (long long)blockIdx.x * blockDim.x + threadIdx.x;
    const long long stride = (long long)gridDim.x * blockDim.x;
    for (; i < n; i += stride) p[i] = 0.0f;
}

// ---------------------------------------------------------------------------
// Y[M,256] = X[M,256] @ W[256,256]   (exact fp32 via V_WMMA_F32_16X16X4_F32)
// fused epilogue: p[m] += dot(Y[m,:], avec)   (p pre-zeroed)
//
// Block = 256 threads = 8 waves, covers one 16-row M band.
// Phase 1: cooperatively stage W *transposed* into LDS (wt[n][k]); the whole
//          256 KB weight matrix fits in the 320 KB WGP LDS. Global reads are
//          fully coalesced float4.
// Phase 2: wave w owns N-tiles {2w, 2w+1}; per k-step the inner loop is
//          1 global b64 (A) + 2 ds_load_b64 (B frags, bank-conflict-free)
//          + 2 v_wmma_f32_16x16x4_f32.
//
// ISA VGPR layouts (CDNA5 ISA):
//   A 16x4 f32 : lane l(<16)/l+16 half h: v2f = A[m0+l][4k+2h], A[m0+l][4k+2h+1]
//   B  4x16    : lane half h supplies rows K=4k+2h,4k+2h+1 at column n0+(l&15)
//   C/D 16x16  : VGPR r, lanes0-15 -> M=r, lanes16-31 -> M=r+8, N = n0+(l&15)
// ---------------------------------------------------------------------------
__global__ __launch_bounds__(256) void HBNS_gemm256_wmma_f32(
    const float* __restrict__ X,     // [M,256] row-major
    const float* __restrict__ W,     // [256,256] row-major
    const float* __restrict__ avec,  // [256]
    float* __restrict__ Y,           // [M,256]
    float* __restrict__ p,           // [M], pre-zeroed
    int M)
{
    extern __shared__ float wt[];    // wt[n * WT_STRIDE + k] = W[k][n]

    const int tid = threadIdx.x;

    // ---- Phase 1: coalesced load + transpose of W into LDS ----
    // 16384 float4 slots over 256 threads = 64 iterations.
    for (int it = 0; it < 64; ++it) {
        const int idx = it * 256 + tid;       // float4 slot in W
        const int k   = idx >> 6;             // W row (K)
        const int n4  = (idx & 63) << 2;      // W column group (N)
        const float4 wv = *(const float4*)(W + k * DHID + n4);
        wt[(n4 + 0) * WT_STRIDE + k] = wv.x;
        wt[(n4 + 1) * WT_STRIDE + k] = wv.y;
        wt[(n4 + 2) * WT_STRIDE + k] = wv.z;
        wt[(n4 + 3) * WT_STRIDE + k] = wv.w;
    }
    __syncthreads();

    // ---- Phase 2: WMMA main loop ----
    const int lane = tid & 31;
    const int wave = tid >> 5;           // 0..7
    const int hl   = lane >> 4;          // 0: lanes 0-15, 1: lanes 16-31
    const int l15  = lane & 15;
    const int m0   = blockIdx.x * 16;

    // Clamp A row so EXEC stays all-ones through the WMMAs (ragged M band).
    const int arow = imin(m0 + l15, M - 1);
    const float* __restrict__ xrow = X + (long long)arow * DHID;

    const int n0 = wave * 32;            // first of two adjacent 16-col tiles
    const int n1 = n0 + 16;
    const float* __restrict__ wt0 = wt + (n0 + l15) * WT_STRIDE;
    const float* __restrict__ wt1 = wt + (n1 + l15) * WT_STRIDE;

    v8f c0 = {};
    v8f c1 = {};

#pragma unroll 8
    for (int kb = 0; kb < 64; ++kb) {
        const int k0 = kb * 4 + hl * 2;
        const v2f av = *(const v2f*)(xrow + k0);   // global b64
        const v2f b0 = *(const v2f*)(wt0 + k0);    // ds_load_b64, conflict-free
        const v2f b1 = *(const v2f*)(wt1 + k0);    // ds_load_b64, conflict-free
        c0 = __builtin_amdgcn_wmma_f32_16x16x4_f32(false, av, false, b0,
                                                   (short)0, c0, false, false);
        c1 = __builtin_amdgcn_wmma_f32_16x16x4_f32(false, av, false, b1,
                                                   (short)0, c1, false, false);
    }

    const float a0 = avec[n0 + l15];
    const float a1 = avec[n1 + l15];

#pragma unroll
    for (int r = 0; r < 8; ++r) {
        const int row = m0 + hl * 8 + r;
        const bool ok = row < M;
        if (ok) {
            Y[(long long)row * DHID + n0 + l15] = c0[r];
            Y[(long long)row * DHID + n1 + l15] = c1[r];
        }
        // fused p[row] += sum_n Y[row,n]*a[n] over this wave's 32 columns:
        // reduce across the 16 lanes that share this row.
        float v = c0[r] * a0 + c1[r] * a1;
        v += __shfl_xor(v, 1, 32);
        v += __shfl_xor(v, 2, 32);
        v += __shfl_xor(v, 4, 32);
        v += __shfl_xor(v, 8, 32);
        if (l15 == 0 && ok) atomicAdd(&p[row], v);
    }
}

// ---------------------------------------------------------------------------
// Edge pass 1: ev = leaky_relu(p_s[col] + p_t[row]); rowsums via f32 atomics
// ---------------------------------------------------------------------------
__global__ __launch_bounds__(256) void HBNS_edge_pass1(
    const int* __restrict__ erow, const int* __restrict__ ecol,
    const float* __restrict__ p_s, const float* __restrict__ p_t,
    float* __restrict__ ev, float* __restrict__ rowsum_s,
    float* __restrict__ rowsum_t, int nE)
{
    const int e = blockIdx.x * blockDim.x + threadIdx.x;
    if (e >= nE) return;
    const int r = erow[e];
    const int c = ecol[e];
    const float x = p_s[c] + p_t[r];
    const float v = (x >= 0.0f) ? x : NEG_SLOPE * x;
    ev[e] = v;
    atomicAdd(&rowsum_t[r], v);
    atomicAdd(&rowsum_s[c], v);
}

// ---------------------------------------------------------------------------
// Edge pass 2: one wave per edge. Lanes cover the 256-float row as 64 float4s
// (coalesced gathers hit L2 since s_msg/t_msg fit in 192 MB); scatter via
// f32 atomic adds into the two output message tensors.
// ---------------------------------------------------------------------------
__global__ __launch_bounds__(256) void HBNS_edge_pass2(
    const int* __restrict__ erow, const int* __restrict__ ecol,
    const float* __restrict__ nvals, const float* __restrict__ ev,
    const float* __restrict__ rowsum_s, const float* __restrict__ rowsum_t,
    const float* __restrict__ s_msg, const float* __restrict__ t_msg,
    float* __restrict__ msg_src, float* __restrict__ msg_tgt, int nE)
{
    const int lane = threadIdx.x & 31;
    const long long wavesTotal = (long long)gridDim.x * (blockDim.x >> 5);
    const long long w0 = (long long)blockIdx.x * (blockDim.x >> 5) + (threadIdx.x >> 5);

    for (long long e = w0; e < nE; e += wavesTotal) {
        const int r = erow[e];
        const int c = ecol[e];
        const float v  = ev[e];
        const float nv = nvals[e];
        const float we = nv * v / rowsum_t[r];   // -> message_on_target
        const float wf = nv * v / rowsum_s[c];   // -> message_on_source

        const float4* __restrict__ srow = (const float4*)(s_msg + (long long)c * DHID);
        const float4* __restrict__ trow = (const float4*)(t_msg + (long long)r * DHID);
        float* __restrict__ mt = msg_tgt + (long long)r * DHID;
        float* __restrict__ ms = msg_src + (long long)c * DHID;

#pragma unroll
        for (int h = 0; h < 2; ++h) {
            const int q = h * 32 + lane;         // float4 index (64 per row)
            const float4 sv = srow[q];
            const float4 tv = trow[q];
            atomicAdd(&mt[q * 4 + 0], we * sv.x);
            atomicAdd(&mt[q * 4 + 1], we * sv.y);
            atomicAdd(&mt[q * 4 + 2], we * sv.z);
            atomicAdd(&mt[q * 4 + 3], we * sv.w);
            atomicAdd(&ms[q * 4 + 0], wf * tv.x);
            atomicAdd(&ms[q * 4 + 1], wf * tv.y);
            atomicAdd(&ms[q * 4 + 2], wf * tv.z);
            atomicAdd(&ms[q * 4 + 3], wf * tv.w);
        }
    }
}

// ---------------------------------------------------------------------------
// Launcher
// ---------------------------------------------------------------------------
extern "C" void kernel_launch(void* const* d_in, const int* in_sizes, int n_in,
                              void* d_out, int out_size, void* d_ws, size_t ws_size,
                              hipStream_t stream)
{
    const float* x_source = (const float*)d_in[0];
    const float* x_target = (const float*)d_in[1];
    const float* n_vals   = (const float*)d_in[2];
    const float* w_s      = (const float*)d_in[3];
    const float* w_t      = (const float*)d_in[4];
    const float* att      = (const float*)d_in[5];
    const int*   edge_row = (const int*)d_in[6];
    const int*   edge_col = (const int*)d_in[7];

    const int n_s = in_sizes[0] / DHID;   // 100000
    const int n_t = in_sizes[1] / DHID;   // 25000
    const int n_e = in_sizes[2];          // 1600000

    // Workspace layout (floats): s_msg | t_msg | p_s | p_t | rowsum_s | rowsum_t | ev
    float* ws       = (float*)d_ws;
    float* s_msg    = ws;
    float* t_msg    = s_msg + (size_t)n_s * DHID;
    float* p_s      = t_msg + (size_t)n_t * DHID;
    float* p_t      = p_s + n_s;
    float* rowsum_s = p_t + n_t;
    float* rowsum_t = rowsum_s + n_s;
    float* ev       = rowsum_t + n_t;

    // Output: (message_on_source [n_s,256], message_on_target [n_t,256]) flat
    float* msg_src = (float*)d_out;
    float* msg_tgt = msg_src + (size_t)n_s * DHID;

    // 1) zero outputs + scalar accumulators
    HBNS_fill_zero<<<4096, 256, 0, stream>>>(msg_src, (long long)(n_s + n_t) * DHID);
    HBNS_fill_zero<<<256, 256, 0, stream>>>(p_s, (long long)2 * (n_s + n_t));

    // 2) fp32 WMMA GEMMs (W staged transposed in 320KB LDS) + fused logits
    HBNS_gemm256_wmma_f32<<<(n_s + 15) / 16, 256, WT_LDS_BYTES, stream>>>(
        x_source, w_s, att, s_msg, p_s, n_s);
    HBNS_gemm256_wmma_f32<<<(n_t + 15) / 16, 256, WT_LDS_BYTES, stream>>>(
        x_target, w_t, att + DHID, t_msg, p_t, n_t);

    // 3) edge logits + segment row-sums
    HBNS_edge_pass1<<<(n_e + 255) / 256, 256, 0, stream>>>(
        edge_row, edge_col, p_s, p_t, ev, rowsum_s, rowsum_t, n_e);

    // 4) normalized gather-scatter of messages (one wave per edge)
    const int wavesPerBlock = 8;
    int blocks = (n_e + wavesPerBlock - 1) / wavesPerBlock;
    if (blocks > (1 << 20)) blocks = (1 << 20);
    HBNS_edge_pass2<<<blocks, 256, 0, stream>>>(
        edge_row, edge_col, n_vals, ev, rowsum_s, rowsum_t,
        s_msg, t_msg, msg_src, msg_tgt, n_e);
}